// HashRouter_11544872091889
// MI455X (gfx1250) — compile-verified
//
#include <hip/hip_runtime.h>

typedef __attribute__((ext_vector_type(2))) float v2f;
typedef __attribute__((ext_vector_type(4))) float v4f;
typedef __attribute__((ext_vector_type(8))) float v8f;

#define HDIM 2048
#define NEXP 64
#define TOK_PER_BLOCK 128   // 8 waves x 16-token tiles

__global__ __launch_bounds__(256) void hash_router_kernel(
    const float* __restrict__ hs,   // (T, 2048)
    const float* __restrict__ hw,   // (64, 2048)
    const float* __restrict__ hb,   // (64,)
    float* __restrict__ out,        // [T*2 weights | T*2 indices | T*64 logits]
    int T)
{
    const int tid  = threadIdx.x;
    const int wave = tid >> 5;
    const int lane = tid & 31;
    const int m0   = blockIdx.x * TOK_PER_BLOCK + wave * 16;

    const int half = lane >> 4;   // 0 or 1
    const int lr   = lane & 15;

    // Permuted-K scheme: per 8-K chunk, lane loads a contiguous float4 at
    // column k0 + 4*half. Pair {0,1} of the float4 feeds WMMA #1, pair {2,3}
    // feeds WMMA #2. A and B use the same slot->K convention, so each WMMA is
    // an exact partial sum over its 4 K values (reduction order is free).
    const float* __restrict__ aptr = hs + (size_t)(m0 + lr) * HDIM + 4 * half;
    const float* __restrict__ bptr = hw + (size_t)lr * HDIM + 4 * half;

    v8f acc0 = {}, acc1 = {}, acc2 = {}, acc3 = {};

    #pragma unroll 2
    for (int k0 = 0; k0 < HDIM; k0 += 8) {
        v4f a  = *(const v4f*)(aptr + k0);
        v4f b0 = *(const v4f*)(bptr + k0);
        v4f b1 = *(const v4f*)(bptr + (size_t)16 * HDIM + k0);
        v4f b2 = *(const v4f*)(bptr + (size_t)32 * HDIM + k0);
        v4f b3 = *(const v4f*)(bptr + (size_t)48 * HDIM + k0);

        // Prefetch the A stream ~1KB ahead (cold HBM stream; B lives in L2).
        __builtin_prefetch(aptr + k0 + 256, 0, 3);

        v2f alo = {a.x, a.y}, ahi = {a.z, a.w};
        v2f b0lo = {b0.x, b0.y}, b0hi = {b0.z, b0.w};
        v2f b1lo = {b1.x, b1.y}, b1hi = {b1.z, b1.w};
        v2f b2lo = {b2.x, b2.y}, b2hi = {b2.z, b2.w};
        v2f b3lo = {b3.x, b3.y}, b3hi = {b3.z, b3.w};

        acc0 = __builtin_amdgcn_wmma_f32_16x16x4_f32(false, alo, false, b0lo, (short)0, acc0, false, false);
        acc1 = __builtin_amdgcn_wmma_f32_16x16x4_f32(false, alo, false, b1lo, (short)0, acc1, false, false);
        acc2 = __builtin_amdgcn_wmma_f32_16x16x4_f32(false, alo, false, b2lo, (short)0, acc2, false, false);
        acc3 = __builtin_amdgcn_wmma_f32_16x16x4_f32(false, alo, false, b3lo, (short)0, acc3, false, false);
        acc0 = __builtin_amdgcn_wmma_f32_16x16x4_f32(false, ahi, false, b0hi, (short)0, acc0, false, false);
        acc1 = __builtin_amdgcn_wmma_f32_16x16x4_f32(false, ahi, false, b1hi, (short)0, acc1, false, false);
        acc2 = __builtin_amdgcn_wmma_f32_16x16x4_f32(false, ahi, false, b2hi, (short)0, acc2, false, false);
        acc3 = __builtin_amdgcn_wmma_f32_16x16x4_f32(false, ahi, false, b3hi, (short)0, acc3, false, false);
    }

    // Bias: N depends only on lane&15 within each N-tile.
    const float bb0 = hb[ 0 + lr];
    const float bb1 = hb[16 + lr];
    const float bb2 = hb[32 + lr];
    const float bb3 = hb[48 + lr];

    // Sign-bit masks via wave ballot.
    // D layout: VGPR v holds rows M=v (lanes 0-15) and M=v+8 (lanes 16-31), N = lane&15.
    unsigned msk[4][8];
    #pragma unroll
    for (int v = 0; v < 8; ++v) {
        msk[0][v] = __builtin_amdgcn_ballot_w32((acc0[v] + bb0) > 0.0f);
        msk[1][v] = __builtin_amdgcn_ballot_w32((acc1[v] + bb1) > 0.0f);
        msk[2][v] = __builtin_amdgcn_ballot_w32((acc2[v] + bb2) > 0.0f);
        msk[3][v] = __builtin_amdgcn_ballot_w32((acc3[v] + bb3) > 0.0f);
    }

    float* __restrict__ ow = out;                    // (T, 2) weights
    float* __restrict__ oi = out + (size_t)T * 2;    // (T, 2) indices (as float)
    float* __restrict__ ol = out + (size_t)T * 4;    // (T, 64) logits = 0

    if (lane < 16) {
        const int r   = lane;          // row within this wave's 16-token tile
        const int v   = r & 7;
        const int sh  = (r >> 3) * 16; // rows 8..15 live in the high 16 ballot bits
        const unsigned b0 = (msk[0][v] >> sh) & 0xFFFFu;
        const unsigned b1 = (msk[1][v] >> sh) & 0xFFFFu;
        const unsigned b2 = (msk[2][v] >> sh) & 0xFFFFu;
        const unsigned b3 = (msk[3][v] >> sh) & 0xFFFFu;

        int e0 = (__popc(b0) + __popc(b1)) & (NEXP - 1);  // cand0 % 64
        int e1 = (__popc(b2) + __popc(b3)) & (NEXP - 1);  // cand1 % 64
        if (e1 == e0) e1 = (e1 + 1) & (NEXP - 1);         // linear probe, 1 collision max

        const size_t t = (size_t)(m0 + r);
        v2f w = {0.5f, 0.5f};
        v2f e = {(float)e0, (float)e1};
        *(v2f*)(ow + t * 2) = w;
        *(v2f*)(oi + t * 2) = e;
    }

    // Zero router_logits for this block's 128 tokens (128*64 floats = 2048 v4f).
    {
        v4f* __restrict__ z = (v4f*)(ol + (size_t)blockIdx.x * TOK_PER_BLOCK * NEXP);
        #pragma unroll
        for (int i = 0; i < (TOK_PER_BLOCK * NEXP / 4) / 256; ++i) {
            v4f zero = {};
            z[i * 256 + tid] = zero;
        }
    }
}

extern "C" void kernel_launch(void* const* d_in, const int* in_sizes, int n_in,
                              void* d_out, int out_size, void* d_ws, size_t ws_size,
                              hipStream_t stream) {
    const float* hs = (const float*)d_in[0];   // (4,4096,2048) fp32
    const float* hw = (const float*)d_in[1];   // (64,2048) fp32
    const float* hb = (const float*)d_in[2];   // (64,) fp32
    float* out = (float*)d_out;

    const int T = in_sizes[0] / HDIM;          // 16384
    const int blocks = T / TOK_PER_BLOCK;      // 128

    hash_router_kernel<<<blocks, 256, 0, stream>>>(hs, hw, hb, out, T);
}